// LinearEncoder_12876311953980
// MI455X (gfx1250) — compile-verified
//
#include <hip/hip_runtime.h>

typedef __attribute__((ext_vector_type(2))) float v2f;
typedef __attribute__((ext_vector_type(8))) float v8f;

#define V_DIM 100000
#define E_DIM 128
#define B_DIM 4096
#define TILE_M 64
#define LDS_STRIDE 132   // 132 % 64 == 4 -> rows hit distinct banks for lanes 0..15

// Fused: gather (one_hot @ W0.T == column-gather of W0) -> GEMM1(+b1) -> GEMM2(+b2)
// All matrix math on V_WMMA_F32_16X16X4_F32 (full fp32, matches reference dtype).
__global__ __launch_bounds__(256) void encoder_mlp_kernel(
    const int*   __restrict__ user_ids,   // [B]
    const float* __restrict__ W0,         // [E, V] row-major
    const float* __restrict__ W1,         // [E, E] row-major (out, in)
    const float* __restrict__ b1,         // [E]
    const float* __restrict__ W2,         // [E, E]
    const float* __restrict__ b2,         // [E]
    float*       __restrict__ out)        // [B, E]
{
    __shared__ float h0[TILE_M * LDS_STRIDE];
    __shared__ float h1[TILE_M * LDS_STRIDE];
    __shared__ int   ids[TILE_M];

    const int tid  = threadIdx.x;
    const int lane = tid & 31;
    const int wave = tid >> 5;            // 0..7
    const int row0 = blockIdx.x * TILE_M; // first batch row of this block

    if (tid < TILE_M) ids[tid] = user_ids[row0 + tid];
    __syncthreads();

    // ---- Phase 1: gather h0[r][e] = W0[e*V + ids[r]]  (random 4B reads, L2-resident W0)
    for (int i = tid; i < TILE_M * E_DIM; i += 256) {
        const int r = i >> 7;       // / 128
        const int e = i & 127;      // % 128
        h0[r * LDS_STRIDE + e] = W0[(long)e * V_DIM + ids[r]];
    }
    __syncthreads();

    // WMMA f32 16x16x4 lane mapping (wave32):
    //   A:   M = lane%16,            K = 2*(lane/16) + vgpr
    //   B:   N = lane%16,            K = 2*(lane/16) + vgpr
    //   C/D: M = vgpr + 8*(lane/16), N = lane%16
    const int lg  = lane >> 4;      // lane group 0/1
    const int lm  = lane & 15;      // row (A) / col (B,C) within tile
    const int mt  = wave & 3;       // M tile 0..3  (16 rows each)
    const int nt0 = (wave >> 2) * 4;// N tile base 0 or 4 (this wave does 4 N tiles)

    // ---- Phase 2: h1 = h0 @ W1^T + b1 ----
    {
        v8f acc[4];
        #pragma unroll
        for (int j = 0; j < 4; ++j) {
            const float bv = b1[(nt0 + j) * 16 + lm];
            #pragma unroll
            for (int v = 0; v < 8; ++v) acc[j][v] = bv;
        }
        const float* arow = &h0[(mt * 16 + lm) * LDS_STRIDE + lg * 2];
        #pragma unroll 4
        for (int k0 = 0; k0 < E_DIM; k0 += 4) {
            const v2f a = *(const v2f*)(arow + k0);
            #pragma unroll
            for (int j = 0; j < 4; ++j) {
                const v2f b = *(const v2f*)(&W1[((nt0 + j) * 16 + lm) * E_DIM + k0 + lg * 2]);
                acc[j] = __builtin_amdgcn_wmma_f32_16x16x4_f32(
                    false, a, false, b, (short)0, acc[j], false, false);
            }
        }
        #pragma unroll
        for (int j = 0; j < 4; ++j) {
            const int n = (nt0 + j) * 16 + lm;
            #pragma unroll
            for (int v = 0; v < 8; ++v) {
                const int m = mt * 16 + v + 8 * lg;
                h1[m * LDS_STRIDE + n] = acc[j][v];
            }
        }
    }
    __syncthreads();

    // ---- Phase 3: out = h1 @ W2^T + b2 ----
    {
        v8f acc[4];
        #pragma unroll
        for (int j = 0; j < 4; ++j) {
            const float bv = b2[(nt0 + j) * 16 + lm];
            #pragma unroll
            for (int v = 0; v < 8; ++v) acc[j][v] = bv;
        }
        const float* arow = &h1[(mt * 16 + lm) * LDS_STRIDE + lg * 2];
        #pragma unroll 4
        for (int k0 = 0; k0 < E_DIM; k0 += 4) {
            const v2f a = *(const v2f*)(arow + k0);
            #pragma unroll
            for (int j = 0; j < 4; ++j) {
                const v2f b = *(const v2f*)(&W2[((nt0 + j) * 16 + lm) * E_DIM + k0 + lg * 2]);
                acc[j] = __builtin_amdgcn_wmma_f32_16x16x4_f32(
                    false, a, false, b, (short)0, acc[j], false, false);
            }
        }
        #pragma unroll
        for (int j = 0; j < 4; ++j) {
            const int n = (nt0 + j) * 16 + lm;
            #pragma unroll
            for (int v = 0; v < 8; ++v) {
                const int m = row0 + mt * 16 + v + 8 * lg;
                out[m * E_DIM + n] = acc[j][v];
            }
        }
    }
}

extern "C" void kernel_launch(void* const* d_in, const int* in_sizes, int n_in,
                              void* d_out, int out_size, void* d_ws, size_t ws_size,
                              hipStream_t stream) {
    const int*   user_ids = (const int*)  d_in[0];
    const float* W0       = (const float*)d_in[1];
    const float* W1       = (const float*)d_in[2];
    const float* b1       = (const float*)d_in[3];
    const float* W2       = (const float*)d_in[4];
    const float* b2       = (const float*)d_in[5];
    float*       out      = (float*)d_out;

    const int grid = B_DIM / TILE_M;  // 64 blocks
    encoder_mlp_kernel<<<grid, 256, 0, stream>>>(user_ids, W0, W1, b1, W2, b2, out);
    (void)in_sizes; (void)n_in; (void)out_size; (void)d_ws; (void)ws_size;
}